// OCGNN_64948495450714
// MI455X (gfx1250) — compile-verified
//
#include <hip/hip_runtime.h>

typedef __attribute__((ext_vector_type(2))) float v2f;
typedef __attribute__((ext_vector_type(8))) float v8f;

#define FEAT 128

// -------------------------------------------------------------------------
// zero fill (workspace is poisoned with 0xAA by the harness)
// -------------------------------------------------------------------------
__global__ void ocgnn_zero_f32(float* __restrict__ p, long long n) {
    long long i = (long long)blockIdx.x * blockDim.x + threadIdx.x;
    long long stride = (long long)gridDim.x * blockDim.x;
    for (; i < n; i += stride) p[i] = 0.0f;
}

// -------------------------------------------------------------------------
// degree accumulation: out_deg[src[e]] += 1 ; in_deg[dst[e]] += 1
// -------------------------------------------------------------------------
__global__ void ocgnn_degrees(const int* __restrict__ src, const int* __restrict__ dst,
                              float* __restrict__ odeg, float* __restrict__ ideg, int E) {
    int i = blockIdx.x * blockDim.x + threadIdx.x;
    if (i >= E) return;
    __hip_atomic_fetch_add(&odeg[src[i]], 1.0f, __ATOMIC_RELAXED, __HIP_MEMORY_SCOPE_AGENT);
    __hip_atomic_fetch_add(&ideg[dst[i]], 1.0f, __ATOMIC_RELAXED, __HIP_MEMORY_SCOPE_AGENT);
}

// deg -> rsqrt(max(deg, 1)) in place
__global__ void ocgnn_to_norm(float* __restrict__ deg, int n) {
    int i = blockIdx.x * blockDim.x + threadIdx.x;
    if (i < n) deg[i] = rsqrtf(fmaxf(deg[i], 1.0f));
}

// -------------------------------------------------------------------------
// SpMM scatter: agg[dst[e], :] += x[src[e], :] * onorm[src[e]]
// One wave per edge (32 lanes x 4 floats). Edge metadata is wave-uniform ->
// readfirstlane to SGPRs so addresses become saddr + lane offset.
// Hardware f32 atomic add (global_atomic_add_f32).
// -------------------------------------------------------------------------
__global__ void ocgnn_spmm(const float* __restrict__ x, const float* __restrict__ onorm,
                           const int* __restrict__ src, const int* __restrict__ dst,
                           float* __restrict__ agg, int E) {
    long long gid = (long long)blockIdx.x * blockDim.x + threadIdx.x;
    if (gid >= (long long)E * 32) return;
    const int e   = (int)(gid >> 5);
    const int col = ((int)gid & 31) * 4;
    const int s = __builtin_amdgcn_readfirstlane(src[e]);
    const int d = __builtin_amdgcn_readfirstlane(dst[e]);
    const float sc = __builtin_amdgcn_readfirstlane(__float_as_uint(onorm[s])) == 0
                         ? 0.0f
                         : onorm[s];  // keep simple: vector load is fine, see below
    const float4 v = *(const float4*)(x + (size_t)s * FEAT + col);
    float* o = agg + (size_t)d * FEAT + col;
    __hip_atomic_fetch_add(o + 0, v.x * sc, __ATOMIC_RELAXED, __HIP_MEMORY_SCOPE_AGENT);
    __hip_atomic_fetch_add(o + 1, v.y * sc, __ATOMIC_RELAXED, __HIP_MEMORY_SCOPE_AGENT);
    __hip_atomic_fetch_add(o + 2, v.z * sc, __ATOMIC_RELAXED, __HIP_MEMORY_SCOPE_AGENT);
    __hip_atomic_fetch_add(o + 3, v.w * sc, __ATOMIC_RELAXED, __HIP_MEMORY_SCOPE_AGENT);
}

// -------------------------------------------------------------------------
// Dense GEMM with WMMA f32 16x16x4:
//   Out[r, :] = maybe_relu( (Ain[r, :] * innorm[r]) @ W )     K = 128 fixed
//
// W [128, NC] row-major is staged into LDS *pair-swizzled*:
//   lds_pair[kp*NC + n] = ( W[2kp][n], W[2kp+1][n] )   (v2f, 8B aligned)
// so every B fragment is one ds_load_b64 into an adjacent VGPR pair —
// exactly the 4x16 f32 B layout (lane half h covers K in {2h,2h+1}).
// A frag (16x4 f32): lane l -> m=l&15, h=l>>4; (A[m][k0+2h], A[m][k0+2h+1]).
// C/D: VGPR v, lane l -> M = v + 8h, N = l&15.
// One wave per 16-row tile, 8 waves per block; A loads are software-
// pipelined one k-step ahead of the WMMA chain.
// -------------------------------------------------------------------------
template <int NT, bool RELU>
__global__ __launch_bounds__(256)
void ocgnn_gemm_wmma(const float* __restrict__ Ain, const float* __restrict__ innorm,
                     const float* __restrict__ W, float* __restrict__ Out, int nrows) {
    constexpr int NC = NT * 16;
    __shared__ v2f wpair[(FEAT / 2) * NC];   // 64 * NC v2f = same 4*K*NC bytes

    // cooperative pair-swizzle stage of W into LDS
    for (int i = threadIdx.x; i < (FEAT / 2) * NC; i += 256) {
        const int kp = i / NC;
        const int n  = i - kp * NC;
        v2f p;
        p.x = W[(2 * kp)     * NC + n];
        p.y = W[(2 * kp + 1) * NC + n];
        wpair[i] = p;
    }
    __syncthreads();

    const int wave = threadIdx.x >> 5;
    const int lane = threadIdx.x & 31;
    const int tile = blockIdx.x * 8 + wave;
    const int row0 = tile * 16;
    if (row0 >= nrows) return;   // whole-wave uniform: EXEC stays all-ones

    const int m = lane & 15;
    const int h = lane >> 4;
    const int row = row0 + m;
    const float s = innorm[row];
    const float* arow = Ain + (size_t)row * FEAT + 2 * h;

    v8f acc[NT] = {};

    // software-pipelined A fragment
    v2f araw = *(const v2f*)(arow);

#pragma unroll 4
    for (int k0 = 0; k0 < FEAT; k0 += 4) {
        v2f a;
        a.x = araw.x * s;
        a.y = araw.y * s;
        if (k0 + 4 < FEAT) araw = *(const v2f*)(arow + k0 + 4);

        const v2f* brow = wpair + ((k0 >> 1) + h) * NC + m;
#pragma unroll
        for (int t = 0; t < NT; ++t) {
            const v2f b = brow[t * 16];
            acc[t] = __builtin_amdgcn_wmma_f32_16x16x4_f32(
                false, a, false, b, (short)0, acc[t], false, false);
        }
    }

#pragma unroll
    for (int t = 0; t < NT; ++t) {
#pragma unroll
        for (int v = 0; v < 8; ++v) {
            float val = acc[t][v];
            if (RELU) val = fmaxf(val, 0.0f);
            Out[(size_t)(row0 + v + 8 * h) * NC + t * 16 + m] = val;
        }
    }
}

// -------------------------------------------------------------------------
// Host-side orchestration
// -------------------------------------------------------------------------
extern "C" void kernel_launch(void* const* d_in, const int* in_sizes, int n_in,
                              void* d_out, int out_size, void* d_ws, size_t ws_size,
                              hipStream_t stream) {
    const float* feat = (const float*)d_in[0];  // [N,128]
    const float* W1   = (const float*)d_in[1];  // [128,128]
    const float* W2   = (const float*)d_in[2];  // [128,64]
    const int*   src  = (const int*)d_in[3];    // [E]
    const int*   dst  = (const int*)d_in[4];    // [E]

    const int N = in_sizes[0] / FEAT;
    const int E = in_sizes[3];

    float* out = (float*)d_out;                 // [N,64]

    // workspace layout (floats): onorm[N] | inorm[N] | agg[N*128] | h1[N*128]
    float* ws    = (float*)d_ws;
    float* onorm = ws;
    float* inorm = ws + N;
    float* agg   = ws + 2 * (size_t)N;
    float* h1    = agg + (size_t)N * FEAT;

    const long long aggN = (long long)N * FEAT;
    const int ZB = 1024;  // blocks for grid-stride zero kernels

    // ---- degree norms ----
    ocgnn_zero_f32<<<ZB, 256, 0, stream>>>(onorm, 2LL * N);
    ocgnn_degrees<<<(E + 255) / 256, 256, 0, stream>>>(src, dst, onorm, inorm, E);
    ocgnn_to_norm<<<(2 * N + 255) / 256, 256, 0, stream>>>(onorm, 2 * N);

    const int tiles   = (N + 15) / 16;
    const int gblocks = (tiles + 7) / 8;
    const long long spmm_threads = (long long)E * 32;
    const int spmm_blocks = (int)((spmm_threads + 255) / 256);

    // ---- layer 1: agg = A @ scaled(feat); h1 = relu((agg*in_norm) @ W1) ----
    ocgnn_zero_f32<<<ZB, 256, 0, stream>>>(agg, aggN);
    ocgnn_spmm<<<spmm_blocks, 256, 0, stream>>>(feat, onorm, src, dst, agg, E);
    ocgnn_gemm_wmma<8, true><<<gblocks, 256, 0, stream>>>(agg, inorm, W1, h1, N);

    // ---- layer 2: reuse agg buffer; out = (agg*in_norm) @ W2 ----
    ocgnn_zero_f32<<<ZB, 256, 0, stream>>>(agg, aggN);
    ocgnn_spmm<<<spmm_blocks, 256, 0, stream>>>(h1, onorm, src, dst, agg, E);
    ocgnn_gemm_wmma<4, false><<<gblocks, 256, 0, stream>>>(agg, inorm, W2, out, N);
}